// ProximityAwareLoss4Class_54674933678210
// MI455X (gfx1250) — compile-verified
//
#include <hip/hip_runtime.h>
#include <hip/hip_bf16.h>
#include <stdint.h>

#define TOL        5
#define SSIZE      8192
#define NWORDS     (SSIZE / 32)         // 256 mask words per row
#define NTHREADS   256
#define PER_THREAD (SSIZE / NTHREADS)   // 32 positions per thread

// ---- CDNA5 async global->LDS path (gfx1250), guarded so we always compile ----
#if defined(__gfx1250__) && __has_builtin(__builtin_amdgcn_global_load_async_to_lds_b128) && __has_builtin(__builtin_amdgcn_s_wait_asynccnt)
#define USE_ASYNC_LDS 1
#else
#define USE_ASYNC_LDS 0
#endif

// The async-to-LDS b128 builtin takes pointers to a 16-byte int vector in
// the global (addrspace 1) and LDS (addrspace 3) address spaces.
typedef int v4i __attribute__((vector_size(4 * sizeof(int))));
typedef __attribute__((address_space(1))) v4i g_v4i;
typedef __attribute__((address_space(3))) v4i l_v4i;

__global__ __launch_bounds__(NTHREADS) void pal_row_kernel(
    const float* __restrict__ logits,   // [B, S, 4]
    const int*   __restrict__ labels,   // [B, S]
    const float* __restrict__ cw,       // [4]
    float* __restrict__ psum,           // [B] per-row weighted-CE sums
    float* __restrict__ pcnt)           // [B] per-row valid counts
{
    __shared__ int          s_lab[SSIZE];      // 32 KB (async DMA target)
    __shared__ unsigned int s_tm2[NWORDS];     // label==2 bitmask, 1 KB
    __shared__ unsigned int s_tm3[NWORDS];     // label==3 bitmask
    __shared__ unsigned int s_pm2[NWORDS];     // pred==2 bitmask
    __shared__ unsigned int s_pm3[NWORDS];     // pred==3 bitmask
    __shared__ int          s_any[2];
    __shared__ float        s_rsum[NTHREADS];
    __shared__ float        s_rcnt[NTHREADS];

    const int tid = threadIdx.x;
    const int row = blockIdx.x;
    const long rowBase = (long)row * SSIZE;

    const int*   labRow = labels + rowBase;
    const float* logRow = logits + rowBase * 4;

    if (tid < 2) s_any[tid] = 0;

    // ---- stage label row into LDS via CDNA5 async DMA (no VGPR round trip) ----
#if USE_ASYNC_LDS
    #pragma unroll
    for (int i = 0; i < PER_THREAD / 4; ++i) {          // 8 x B128 = 128 B/thread
        int e = (tid + i * NTHREADS) * 4;               // int element index
        __builtin_amdgcn_global_load_async_to_lds_b128(
            (g_v4i*)(labRow + e),
            (l_v4i*)(s_lab + e),
            0, 0);
    }
#else
    #pragma unroll
    for (int i = 0; i < PER_THREAD; ++i) {
        int s = tid + i * NTHREADS;
        s_lab[s] = labRow[s];
    }
#endif

    // broadcast class weights while the async copy is in flight
    const float w0 = cw[0], w1 = cw[1], w2 = cw[2], w3 = cw[3];

#if USE_ASYNC_LDS
    __builtin_amdgcn_s_wait_asynccnt(0);
#endif
    __syncthreads();

    // ---- phase 1: log-softmax CE + argmax; build row bitmasks via wave32 ballots ----
    // s = tid + i*256: each wave's 32 lanes cover 32 consecutive positions, so
    // one __ballot == one mask word at index (waveId + i*8).
    float ce_reg[PER_THREAD];
    int   lcnt = 0;
    int   anyLocal2 = 0, anyLocal3 = 0;
    const int lane0 = ((tid & 31) == 0);
    const int wid   = tid >> 5;
    const float4* logRow4 = (const float4*)logRow;

    #pragma unroll 4
    for (int i = 0; i < PER_THREAD; ++i) {
        int s = tid + i * NTHREADS;                     // coalesced float4 per lane
        float4 v = logRow4[s];

        int amax = 0; float best = v.x;                 // first-occurrence argmax
        if (v.y > best) { best = v.y; amax = 1; }
        if (v.z > best) { best = v.z; amax = 2; }
        if (v.w > best) { best = v.w; amax = 3; }
        float lse = best + __logf(__expf(v.x - best) + __expf(v.y - best) +
                                  __expf(v.z - best) + __expf(v.w - best));

        int lab  = s_lab[s];
        int safe = lab < 0 ? 0 : lab;
        float xv = (safe == 0) ? v.x : (safe == 1) ? v.y : (safe == 2) ? v.z : v.w;
        float ws = (safe == 0) ? w0  : (safe == 1) ? w1  : (safe == 2) ? w2  : w3;
        ce_reg[i] = (lab < 0) ? 0.0f : ws * (lse - xv);
        lcnt += (lab >= 0);

        unsigned bt2 = (unsigned)__ballot(lab  == 2);
        unsigned bt3 = (unsigned)__ballot(lab  == 3);
        unsigned bp2 = (unsigned)__ballot(amax == 2);
        unsigned bp3 = (unsigned)__ballot(amax == 3);
        anyLocal2 |= (bp2 != 0u);
        anyLocal3 |= (bp3 != 0u);
        if (lane0) {
            int widx = wid + i * 8;
            s_tm2[widx] = bt2; s_tm3[widx] = bt3;
            s_pm2[widx] = bp2; s_pm3[widx] = bp3;
        }
    }
    if (anyLocal2) atomicOr(&s_any[0], 1);
    if (anyLocal3) atomicOr(&s_any[1], 1);
    __syncthreads();

    // ---- phase 2: branchless proximity factors from bitmask windows ----
    const int any2 = s_any[0], any3 = s_any[1];
    float lsum = 0.0f;

    #pragma unroll 2
    for (int i = 0; i < PER_THREAD; ++i) {
        int s = tid + i * NTHREADS;
        int k = s >> 5, p = s & 31;

        unsigned tm2w = s_tm2[k], tm3w = s_tm3[k];
        unsigned pm2w = s_pm2[k], pm3w = s_pm3[k];
        int lab2 = (tm2w >> p) & 1, lab3 = (tm3w >> p) & 1;
        int prd2 = (pm2w >> p) & 1, prd3 = (pm3w >> p) & 1;

        // --- decay: pred==c, lab!=c, nearest true-switch of class c at d in [1,TOL]
        const unsigned* tsel = prd3 ? s_tm3 : s_tm2;    // class picked by prediction
        unsigned tw_c = prd3 ? tm3w : tm2w;
        unsigned tw_l = (k > 0)          ? tsel[k - 1] : 0u;
        unsigned tw_r = (k < NWORDS - 1) ? tsel[k + 1] : 0u;
        uint64_t tv_hi = ((uint64_t)tw_r << 32) | tw_c; // bit j <-> pos k*32+j
        uint64_t tv_lo = ((uint64_t)tw_c << 32) | tw_l; // bit (p+32) <-> pos s
        unsigned rm = (unsigned)(tv_hi >> (p + 1)) & 0x1Fu;   // right: dist 1..5
        unsigned lm = (unsigned)(tv_lo >> (p + 27)) & 0x1Fu;  // left: bit j <-> dist 5-j
        int dr = rm ? __ffs(rm)       : 99;
        int dl = lm ? (__clz(lm) - 26) : 99;
        int d  = dr < dl ? dr : dl;
        int doDecay = (((prd2 & (lab2 ^ 1)) | (prd3 & (lab3 ^ 1))) != 0) & (d <= TOL);
        float decay = doDecay
            ? ((d == 1) ? 0.8f : (d == 2) ? 0.64f : (d == 3) ? 0.512f
                               : (d == 4) ? 0.4096f : 0.32768f)
            : 1.0f;

        // --- tfac: lab==c; x2 if no pred of class c in row, x1.5 if nearest pred > TOL
        const unsigned* psel = lab3 ? s_pm3 : s_pm2;    // class picked by label
        unsigned pw_c = lab3 ? pm3w : pm2w;
        unsigned pw_l = (k > 0)          ? psel[k - 1] : 0u;
        unsigned pw_r = (k < NWORDS - 1) ? psel[k + 1] : 0u;
        uint64_t pv_hi = ((uint64_t)pw_r << 32) | pw_c;
        uint64_t pv_lo = ((uint64_t)pw_c << 32) | pw_l;
        unsigned prm = (unsigned)(pv_hi >> p) & 0x3Fu;        // dist 0..5 (incl self)
        unsigned plm = (unsigned)(pv_lo >> (p + 27)) & 0x1Fu; // dist 1..5 left
        int found = ((prm | plm) != 0u);
        int anyC  = lab3 ? any3 : any2;
        int isT   = lab2 | lab3;
        float tfac = isT ? (!anyC ? 2.0f : (!found ? 1.5f : 1.0f)) : 1.0f;

        lsum += ce_reg[i] * decay * tfac;
    }

    // ---- deterministic block tree reduction ----
    s_rsum[tid] = lsum;
    s_rcnt[tid] = (float)lcnt;
    __syncthreads();
    for (int off = NTHREADS / 2; off > 0; off >>= 1) {
        if (tid < off) {
            s_rsum[tid] += s_rsum[tid + off];
            s_rcnt[tid] += s_rcnt[tid + off];
        }
        __syncthreads();
    }
    if (tid == 0) { psum[row] = s_rsum[0]; pcnt[row] = s_rcnt[0]; }
}

__global__ __launch_bounds__(256) void pal_finalize(
    const float* __restrict__ psum, const float* __restrict__ pcnt,
    float* __restrict__ out, int B)
{
    __shared__ float ss[256], sc[256];
    int tid = threadIdx.x;
    float a = 0.0f, b = 0.0f;
    for (int i = tid; i < B; i += 256) { a += psum[i]; b += pcnt[i]; }
    ss[tid] = a; sc[tid] = b;
    __syncthreads();
    for (int off = 128; off > 0; off >>= 1) {
        if (tid < off) { ss[tid] += ss[tid + off]; sc[tid] += sc[tid + off]; }
        __syncthreads();
    }
    if (tid == 0) out[0] = ss[0] / fmaxf(sc[0], 1.0f);
}

extern "C" void kernel_launch(void* const* d_in, const int* in_sizes, int n_in,
                              void* d_out, int out_size, void* d_ws, size_t ws_size,
                              hipStream_t stream) {
    const float* logits = (const float*)d_in[0];   // [B, S, 4] f32
    const int*   labels = (const int*)d_in[1];     // [B, S]    i32
    const float* cw     = (const float*)d_in[2];   // [4]       f32

    const int B = in_sizes[1] / SSIZE;             // 512 rows

    float* psum = (float*)d_ws;                    // [B]
    float* pcnt = psum + B;                        // [B]

    pal_row_kernel<<<B, NTHREADS, 0, stream>>>(logits, labels, cw, psum, pcnt);
    pal_finalize<<<1, 256, 0, stream>>>(psum, pcnt, (float*)d_out, B);
}